// Model_41266045780566
// MI455X (gfx1250) — compile-verified
//
#include <hip/hip_runtime.h>

// Problem constants (match reference).
constexpr int BATCH   = 32;
constexpr int CHAN    = 16;
constexpr int LEN     = 131072;
constexpr int LOUT    = 65536;      // (LEN + 2*2 - 5)/2 + 1
constexpr int THREADS = 256;
constexpr int OPT     = 8;                    // outputs per thread
constexpr int CHUNK   = THREADS * OPT;        // 2048 outputs per block
constexpr int CHUNKS_PER_ROW = LOUT / CHUNK;  // 32
constexpr int NELEM   = 2 * CHUNK + 3;        // 4099 staged inputs per block

#if __has_builtin(__builtin_amdgcn_global_load_async_to_lds_b32)
#define HAVE_ASYNC_LDS 1
typedef __attribute__((address_space(1))) int gas_int;  // "__device__ int*" per clang diag
typedef __attribute__((address_space(3))) int las_int;  // "__shared__ int*"
#endif

static __device__ __forceinline__ float log_sigmoid(float x) {
  // log_sigmoid(x) = min(x,0) - log1p(exp(-|x|)), via native v_exp_f32 / v_log_f32
  float ax = __builtin_fabsf(x);
  float e  = __builtin_amdgcn_exp2f(ax * -1.44269504088896340736f);     // exp(-|x|)
  float l  = __builtin_amdgcn_logf(1.0f + e) * 0.69314718055994530942f; // ln(1+e)
  return __builtin_fminf(x, 0.0f) - l;
}

static __device__ __forceinline__ float fast_tanh(float z) {
#if __has_builtin(__builtin_amdgcn_tanhf)
  return __builtin_amdgcn_tanhf(z);
#else
  float r;
  // v_nop covers the TRANS-op result hazard for hand-written asm.
  asm volatile("v_tanh_f32 %0, %1\n\tv_nop" : "=v"(r) : "v"(z));
  return r;
#endif
}

__global__ __launch_bounds__(THREADS) void
logsig_conv1d_kernel(const float* __restrict__ x,
                     const float* __restrict__ wgt,   // [CHAN][5]
                     const float* __restrict__ bias,  // [CHAN]
                     float* __restrict__ out) {
  __shared__ __align__(16) float smem[NELEM + 1];  // pad so float4 tail read is in-bounds

  const int t     = threadIdx.x;
  const int row   = blockIdx.x / CHUNKS_PER_ROW;   // (b*CHAN + c)
  const int chunk = blockIdx.x % CHUNKS_PER_ROW;
  const int c     = row % CHAN;
  const float* xrow = x + (size_t)row * LEN;
  const int o0 = chunk * CHUNK;
  const int s  = 2 * o0 - 2;  // first (unclamped) input index needed by this chunk

  // ---- Phase 1: stage clamped inputs into LDS (async global->LDS on CDNA5) ----
#pragma unroll
  for (int r = 0; r < 17; ++r) {
    const int e = r * THREADS + t;
    if (r == 16 && t >= NELEM - 16 * THREADS) break;  // tail: 3 halo elements
    int ip  = s + e;
    int idx = ip < 0 ? 0 : (ip > LEN - 1 ? LEN - 1 : ip);  // edge replication
#if defined(HAVE_ASYNC_LDS)
    __builtin_amdgcn_global_load_async_to_lds_b32(
        (gas_int*)(xrow + idx), (las_int*)(&smem[e]),
        0 /*offset*/, 0 /*cpol*/);
#else
    smem[e] = xrow[idx];
#endif
  }
#if defined(HAVE_ASYNC_LDS)
#if __has_builtin(__builtin_amdgcn_s_wait_asynccnt)
  __builtin_amdgcn_s_wait_asynccnt(0);
#else
  asm volatile("s_wait_asynccnt 0" ::: "memory");
#endif
#endif
  __syncthreads();

  // ---- Phase 2: log_sigmoid in place, each input transformed exactly once ----
  float4* sv = reinterpret_cast<float4*>(smem);
#pragma unroll
  for (int q = 0; q < 4; ++q) {
    float4 v = sv[4 * t + q];
    v.x = log_sigmoid(v.x);
    v.y = log_sigmoid(v.y);
    v.z = log_sigmoid(v.z);
    v.w = log_sigmoid(v.w);
    sv[4 * t + q] = v;
  }
  if (t < NELEM - 16 * THREADS) {  // 3 halo elements
    const int e = 16 * THREADS + t;
    smem[e] = log_sigmoid(smem[e]);
  }
  __syncthreads();

  // ---- Phase 3: 5-tap strided dot + bias + tanh, 8 outputs per thread ----
  const float w0 = wgt[c * 5 + 0], w1 = wgt[c * 5 + 1], w2 = wgt[c * 5 + 2],
              w3 = wgt[c * 5 + 3], w4 = wgt[c * 5 + 4];
  const float bz = bias[c];

  float Lv[20];
#pragma unroll
  for (int q = 0; q < 5; ++q)
    *reinterpret_cast<float4*>(&Lv[4 * q]) = sv[4 * t + q];  // smem[16t .. 16t+19]

  float res[OPT];
#pragma unroll
  for (int j = 0; j < OPT; ++j) {
    float acc = Lv[2 * j] * w0;
    acc = fmaf(Lv[2 * j + 1], w1, acc);
    acc = fmaf(Lv[2 * j + 2], w2, acc);
    acc = fmaf(Lv[2 * j + 3], w3, acc);
    acc = fmaf(Lv[2 * j + 4], w4, acc);
    res[j] = fast_tanh(acc + bz);
  }

  float* orow = out + (size_t)row * LOUT + o0 + OPT * t;
  float4 r0 = {res[0], res[1], res[2], res[3]};
  float4 r1 = {res[4], res[5], res[6], res[7]};
  reinterpret_cast<float4*>(orow)[0] = r0;
  reinterpret_cast<float4*>(orow)[1] = r1;
}

extern "C" void kernel_launch(void* const* d_in, const int* in_sizes, int n_in,
                              void* d_out, int out_size, void* d_ws, size_t ws_size,
                              hipStream_t stream) {
  const float* x    = (const float*)d_in[0];
  const float* wgt  = (const float*)d_in[1];
  const float* bias = (const float*)d_in[2];
  float* out        = (float*)d_out;

  const int blocks = BATCH * CHAN * CHUNKS_PER_ROW;  // 16384
  logsig_conv1d_kernel<<<dim3(blocks), dim3(THREADS), 0, stream>>>(x, wgt, bias, out);
}